// LeapfrogIntegrator_35759897706557
// MI455X (gfx1250) — compile-verified
//
#include <hip/hip_runtime.h>

typedef __attribute__((ext_vector_type(2))) float v2f;
typedef __attribute__((ext_vector_type(8))) float v8f;
typedef __attribute__((ext_vector_type(4))) int i4;

#define DT_STEP 0.01f
#define EPS_F 1e-8f
#define B_DIM 8
#define N_DIM 4096
#define K_DIM 16
#define SITES (B_DIM * N_DIM)
#define WAVES_PER_BLOCK 8
#define NUM_BLOCKS (SITES / WAVES_PER_BLOCK)   // 4096
#define BLOCKS_PER_B (NUM_BLOCKS / B_DIM)      // 512
#define NS_ITERS 15
#define EXP_ORDER 8

#if defined(__AMDGCN__) && __has_builtin(__builtin_amdgcn_global_load_async_to_lds_b128)
#define HAVE_ASYNC_LDS 1
#endif

// DS ops of one wave are in-order (DScnt); this makes cross-lane LDS
// communication within a wave safe against both HW and compiler reordering.
__device__ __forceinline__ void wsync() {
  asm volatile("s_wait_dscnt 0" ::: "memory");
}

__device__ __forceinline__ void async_wait() {
#ifdef HAVE_ASYNC_LDS
  asm volatile("s_wait_asynccnt 0" ::: "memory");
#endif
}

// Stage a 16x16 fp32 tile (1KB) global -> LDS. Prefer the CDNA5 async
// LDS-DMA path (GLOBAL_LOAD_ASYNC_TO_LDS_B128, ASYNCcnt-tracked, no VGPR
// round-trip); fall back to a plain copy if the builtin is unavailable.
__device__ __forceinline__ void stage_tile(const float* __restrict__ g,
                                           float* l, int lane) {
#ifdef HAVE_ASYNC_LDS
  typedef __attribute__((address_space(1))) i4 gi4;  // global int4
  typedef __attribute__((address_space(3))) i4 li4;  // LDS int4
  gi4* sp0 = (gi4*)((const i4*)g + lane);
  gi4* sp1 = (gi4*)((const i4*)g + lane + 32);
  li4* dp0 = (li4*)((i4*)l + lane);
  li4* dp1 = (li4*)((i4*)l + lane + 32);
  __builtin_amdgcn_global_load_async_to_lds_b128(sp0, dp0, 0, 0);
  __builtin_amdgcn_global_load_async_to_lds_b128(sp1, dp1, 0, 0);
#else
  const float4* s4 = (const float4*)g;
  float4* d4 = (float4*)l;
  d4[lane] = s4[lane];
  d4[lane + 32] = s4[lane + 32];
#endif
}

__device__ __forceinline__ float wave_sum(float v) {
#pragma unroll
  for (int off = 16; off > 0; off >>= 1) v += __shfl_xor(v, off, 32);
  return v;
}

// Full 16x16x16 fp32 matmul: 4 chained V_WMMA_F32_16X16X4_F32.
// A,B are row-major 16x16 tiles in LDS. Result in WMMA C layout:
// lane (hi*16+lo) vgpr r holds element (hi*8+r, lo).
__device__ __forceinline__ v8f mm16(const float* A, const float* B, int lo, int hi) {
  v8f c = {};
#pragma unroll
  for (int k0 = 0; k0 < 16; k0 += 4) {
    v2f a, b;
    // A fragment 16x4: lanes0-15 hold K=k0,k0+1 ; lanes16-31 hold K=k0+2,k0+3
    a.x = A[lo * 16 + (k0 + 2 * hi)];
    a.y = A[lo * 16 + (k0 + 2 * hi + 1)];
    // B fragment 4x16: vgpr0 rows k0|k0+2, vgpr1 rows k0+1|k0+3 (half-lane split)
    b.x = B[(k0 + 2 * hi) * 16 + lo];
    b.y = B[(k0 + 2 * hi + 1) * 16 + lo];
    c = __builtin_amdgcn_wmma_f32_16x16x4_f32(false, a, false, b, (short)0, c,
                                              false, false);
  }
  return c;
}

// Store C-layout accumulator to a row-major 16x16 LDS tile.
__device__ __forceinline__ void stC(float* D, v8f c, int lo, int mb) {
#pragma unroll
  for (int r = 0; r < 8; ++r) D[(mb + r) * 16 + lo] = c[r];
}

struct alignas(16) WaveLds {
  float m[6][256];  // six 16x16 fp32 tiles
  float vec[32];    // pi_mu (16 used)
};

__global__ __launch_bounds__(256) void leapfrog_main(
    const float* __restrict__ mu, const float* __restrict__ Sg,
    const float* __restrict__ phi, const float* __restrict__ pimu,
    const float* __restrict__ piSg, const float* __restrict__ piphi,
    const float* __restrict__ Spr, float* __restrict__ out_mu,
    float* __restrict__ out_S, float* __restrict__ out_phi,
    float* __restrict__ ws) {
  __shared__ WaveLds L[WAVES_PER_BLOCK];
  __shared__ float blockT[WAVES_PER_BLOCK];

  const int tid = threadIdx.x;
  const int w = tid >> 5;
  const int lane = tid & 31;
  const int lo = lane & 15;
  const int hi = lane >> 4;
  const int mb = hi * 8;

  const int site = blockIdx.x * WAVES_PER_BLOCK + w;
  const size_t vbase = (size_t)site * K_DIM;
  const size_t mbase = (size_t)site * K_DIM * K_DIM;
  const size_t pbase = (size_t)site * 3;

  float* B0 = L[w].m[0];
  float* B1 = L[w].m[1];
  float* B2 = L[w].m[2];
  float* B3 = L[w].m[3];
  float* B4 = L[w].m[4];
  float* B5 = L[w].m[5];
  float* vec = L[w].vec;

  // ---- stage Sigma->B0, pi_Sigma->B1, Sigma_prior->B2 (async LDS DMA) ----
  stage_tile(Sg + mbase, B0, lane);
  stage_tile(piSg + mbase, B1, lane);
  stage_tile(Spr + mbase, B2, lane);
  if (lane < 16) vec[lane] = pimu[vbase + lane];
  async_wait();
  wsync();

  float tpart = 0.0f;

  // ---- mu_new = mu + DT * Sigma_prior @ pi_mu ; T_mu partial ----
  if (lane < 16) {
    float s = 0.0f;
#pragma unroll
    for (int j = 0; j < 16; ++j) s += B2[lane * 16 + j] * vec[j];
    out_mu[vbase + lane] = mu[vbase + lane] + DT_STEP * s;
    tpart += 0.5f * vec[lane] * s;
  }

  // ---- phi_new = phi + DT * pi_phi ; T_phi partial ----
  if (lane < 3) {
    float pp = piphi[pbase + lane];
    out_phi[pbase + lane] = phi[pbase + lane] + DT_STEP * pp;
    tpart += 0.5f * pp * pp;
  }

  // ---- T_Sigma = trace((piS @ Sigma)^2) ----
  {
    v8f P = mm16(B1, B0, lo, hi);  // piS @ Sigma
    stC(B2, P, lo, mb);            // B2 (Sprior) is free now
    wsync();
    float acc = 0.0f;
#pragma unroll
    for (int r = 0; r < 8; ++r) acc += P[r] * B2[lo * 16 + (mb + r)];  // P[m][n]*P[n][m]
    tpart += acc;
  }

  // ---- trace normalization of Sigma (+eps*I), matches symmetrize + eps*I ----
  float diag = (lane < 16) ? B0[lane * 17] : 0.0f;
  float tr = wave_sum(diag) + 16.0f * EPS_F;
  float inv_tr = 1.0f / tr;

  // ---- Newton-Schulz: Y -> A^{1/2}, Z -> A^{-1/2}, A = sym(Sigma)/tr + eps ----
  float* Yb = B2;
  float* Zb = B3;
  float* Tb = B4;
  float* Sb = B5;
  {
    v8f Y, Z;
#pragma unroll
    for (int r = 0; r < 8; ++r) {
      int m = mb + r;
      float sv = 0.5f * (B0[m * 16 + lo] + B0[lo * 16 + m]);
      if (m == lo) sv += EPS_F;
      Y[r] = sv * inv_tr;
      Z[r] = (m == lo) ? 1.0f : 0.0f;
    }
    stC(Yb, Y, lo, mb);
    stC(Zb, Z, lo, mb);
    wsync();
  }
  for (int it = 0; it < NS_ITERS; ++it) {
    v8f M = mm16(Zb, Yb, lo, hi);  // Z @ Y
    v8f T;
#pragma unroll
    for (int r = 0; r < 8; ++r)
      T[r] = 0.5f * ((((mb + r) == lo) ? 3.0f : 0.0f) - M[r]);
    stC(Tb, T, lo, mb);
    wsync();
    v8f Yn = mm16(Yb, Tb, lo, hi);  // Y @ T
    stC(Sb, Yn, lo, mb);
    v8f Zn = mm16(Tb, Zb, lo, hi);  // T @ Z
    stC(Yb, Zn, lo, mb);            // old Y buffer reused for new Z
    wsync();
    float* t0 = Yb;
    Yb = Sb;  // new Y
    Sb = Zb;  // old Z buffer becomes spare
    Zb = t0;  // new Z
  }
  // Sigma^{1/2} = sqrt(tr) * Y  (kept implicitly via tr scaling below)

  // ---- W = 2*DT*tr * (Y @ piS @ Y), symmetrized ----
  float* Wb = Zb;  // Z no longer needed
  {
    v8f t1 = mm16(Yb, B1, lo, hi);  // Y @ piS
    stC(Tb, t1, lo, mb);
    wsync();
    v8f Wv = mm16(Tb, Yb, lo, hi);  // (Y piS) Y
    float wscale = 2.0f * DT_STEP * tr;
#pragma unroll
    for (int r = 0; r < 8; ++r) Wv[r] *= wscale;
    stC(Wb, Wv, lo, mb);
    wsync();
#pragma unroll
    for (int r = 0; r < 8; ++r) Wv[r] = 0.5f * (Wv[r] + Wb[lo * 16 + (mb + r)]);
    stC(Wb, Wv, lo, mb);
    wsync();
  }

  // ---- E = exp(W) via Horner Taylor: E = I; E = I + (W@E)/j ----
  float* Eb = Sb;
  {
    v8f E;
#pragma unroll
    for (int r = 0; r < 8; ++r) E[r] = ((mb + r) == lo) ? 1.0f : 0.0f;
    stC(Eb, E, lo, mb);
    wsync();
#pragma unroll
    for (int j = EXP_ORDER; j >= 1; --j) {
      v8f M = mm16(Wb, Eb, lo, hi);
      const float inv = 1.0f / (float)j;  // folded to an immediate when unrolled
#pragma unroll
      for (int r = 0; r < 8; ++r)
        E[r] = (((mb + r) == lo) ? 1.0f : 0.0f) + M[r] * inv;
      stC(Eb, E, lo, mb);
      wsync();
    }
  }

  // ---- Sigma_new = tr * (Y @ E @ Y), symmetrize, store ----
  {
    v8f t2 = mm16(Yb, Eb, lo, hi);
    stC(Tb, t2, lo, mb);
    wsync();
    v8f Sn = mm16(Tb, Yb, lo, hi);
#pragma unroll
    for (int r = 0; r < 8; ++r) Sn[r] *= tr;
    stC(Tb, Sn, lo, mb);
    wsync();
#pragma unroll
    for (int r = 0; r < 8; ++r) {
      float sym = 0.5f * (Sn[r] + Tb[lo * 16 + (mb + r)]);
      out_S[mbase + (size_t)(mb + r) * 16 + lo] = sym;
    }
  }

  // ---- kinetic energy reduction: wave -> block -> workspace ----
  float tsite = wave_sum(tpart);
  if (lane == 0) blockT[w] = tsite;
  __syncthreads();
  if (tid == 0) {
    float a = 0.0f;
#pragma unroll
    for (int i = 0; i < WAVES_PER_BLOCK; ++i) a += blockT[i];
    ws[blockIdx.x] = a;
  }
}

// Deterministic second-stage reduction: one block per b, fixed tree order.
__global__ __launch_bounds__(256) void reduce_T(const float* __restrict__ ws,
                                                float* __restrict__ outT) {
  __shared__ float acc[256];
  int t = threadIdx.x;
  int b = blockIdx.x;
  float v = ws[b * BLOCKS_PER_B + t] + ws[b * BLOCKS_PER_B + 256 + t];
  acc[t] = v;
  __syncthreads();
  for (int s = 128; s > 0; s >>= 1) {
    if (t < s) acc[t] += acc[t + s];
    __syncthreads();
  }
  if (t == 0) outT[b] = acc[0];
}

extern "C" void kernel_launch(void* const* d_in, const int* in_sizes, int n_in,
                              void* d_out, int out_size, void* d_ws,
                              size_t ws_size, hipStream_t stream) {
  (void)in_sizes;
  (void)n_in;
  (void)out_size;
  (void)ws_size;
  const float* mu = (const float*)d_in[0];
  const float* Sigma = (const float*)d_in[1];
  const float* phi = (const float*)d_in[2];
  const float* pi_mu = (const float*)d_in[3];
  const float* pi_Sigma = (const float*)d_in[4];
  const float* pi_phi = (const float*)d_in[5];
  const float* Sigma_prior = (const float*)d_in[6];

  float* out = (float*)d_out;
  float* out_mu = out;                                     // B*N*K
  float* out_S = out_mu + (size_t)SITES * K_DIM;           // B*N*K*K
  float* out_phi = out_S + (size_t)SITES * K_DIM * K_DIM;  // B*N*3
  float* out_T = out_phi + (size_t)SITES * 3;              // B
  float* ws = (float*)d_ws;                                // NUM_BLOCKS floats

  leapfrog_main<<<NUM_BLOCKS, 256, 0, stream>>>(
      mu, Sigma, phi, pi_mu, pi_Sigma, pi_phi, Sigma_prior, out_mu, out_S,
      out_phi, ws);
  reduce_T<<<B_DIM, 256, 0, stream>>>(ws, out_T);
}